// SparseMoe_12094627905537
// MI455X (gfx1250) — compile-verified
//
#include <hip/hip_runtime.h>
#include <hip/hip_bf16.h>
#include <math.h>

// Sparse MoE for MI455X (gfx1250, wave32).
// fp32 reference -> native fp32 matrix path V_WMMA_F32_16X16X4_F32.
// Top-2 rewrite: 2*T*H*H = 137 GFLOP instead of dense 8-expert 550 GFLOP.
// W (128 MB) is L2-resident (192 MB); compute-bound on the fp32 WMMA pipe.
// Tiles are staged with GLOBAL_LOAD_ASYNC_TO_LDS_B128 (ASYNCcnt) and
// double-buffered LDS so the next chunk's copy overlaps the WMMA burst.

#define HDIM   2048
#define NEXP   8
#define NTOK   8192   // B*S = 4*2048
#define TM     128
#define TN     128
#define TK     32
#define LDA    (TK + 4)   // padded stride 36 -> conflict-free A-fragment reads

typedef float v2f __attribute__((ext_vector_type(2)));
typedef float v8f __attribute__((ext_vector_type(8)));

// copies 16 bytes from global address sbase + voff (u32 byte offset) to the LDS address in ldsoff, tracked by ASYNCcnt
#define ASYNC_B128(ldsoff, voff, sbase)                                     \
    asm volatile("global_load_async_to_lds_b128 %0, %1, %2"                 \
                 :: "v"(ldsoff), "v"(voff), "s"(sbase) : "memory")

// ---------------------------------------------------------------- utilities
__global__ void zero_counts_kernel(int* counts) {
    if (threadIdx.x < 2 * NEXP) counts[threadIdx.x] = 0;
}

// ---------------------------------------------------------------- router
// 256 threads = 32 tokens x 8 experts. Each thread computes one logit.
__global__ __launch_bounds__(256) void moe_router_kernel(
    const float* __restrict__ x, const float* __restrict__ Wg,
    const float* __restrict__ bg, float* __restrict__ logits_out,
    int* __restrict__ counts, int* __restrict__ lists, float* __restrict__ wl)
{
    __shared__ float sl[32][NEXP];
    const int tid = threadIdx.x;
    const int tl  = tid >> 3;      // local token 0..31
    const int e   = tid & 7;       // expert 0..7
    const int t   = blockIdx.x * 32 + tl;

    const float* xr = x + (size_t)t * HDIM;
    float acc = 0.0f;
    for (int h = 0; h < HDIM; ++h)
        acc = fmaf(xr[h], Wg[h * NEXP + e], acc);
    acc += bg[e];

    sl[tl][e] = acc;
    logits_out[(size_t)t * NEXP + e] = acc;   // second tuple output
    __syncthreads();

    if (e == 0) {
        float p[NEXP];
        float mx = sl[tl][0];
        #pragma unroll
        for (int i = 1; i < NEXP; ++i) mx = fmaxf(mx, sl[tl][i]);
        #pragma unroll
        for (int i = 0; i < NEXP; ++i) p[i] = __expf(sl[tl][i] - mx);
        // softmax normalization cancels in the top-2 renormalized weights.
        int   i0 = 0; float p0 = p[0];
        #pragma unroll
        for (int i = 1; i < NEXP; ++i)
            if (p[i] > p0) { p0 = p[i]; i0 = i; }   // earliest index on ties
        int   i1 = -1; float p1 = -1.0f;
        #pragma unroll
        for (int i = 0; i < NEXP; ++i)
            if (i != i0 && p[i] > p1) { p1 = p[i]; i1 = i; }
        const float inv = 1.0f / (p0 + p1);
        const float w0 = p0 * inv, w1 = p1 * inv;

        // bin token into (slot0, i0) and (slot1, i1)
        int pos0 = atomicAdd(&counts[i0], 1);
        lists[i0 * NTOK + pos0] = t;
        wl   [i0 * NTOK + pos0] = w0;
        int pos1 = atomicAdd(&counts[NEXP + i1], 1);
        lists[(NEXP + i1) * NTOK + pos1] = t;
        wl   [(NEXP + i1) * NTOK + pos1] = w1;
    }
}

// ---------------------------------------------------------------- expert GEMM
// grid = (TN tiles, max TM tiles, NEXP); 256 threads = 8 waves.
// Each wave: 64x32 sub-tile = 4x2 WMMA accumulators of 16x16.
// slot==0: out = w*(xW + b) (plain store, covers every token exactly once)
// slot==1: out += w*(xW + b) (unique writer per element within the pass)
__global__ __launch_bounds__(256) void moe_expert_gemm_kernel(
    const float* __restrict__ x, const float* __restrict__ W,
    const float* __restrict__ b, const int* __restrict__ counts,
    const int* __restrict__ lists, const float* __restrict__ wl,
    float* __restrict__ out, int slot)
{
    __shared__ __align__(16) float sA[2][TM * LDA];
    __shared__ __align__(16) float sB[2][TK * TN];
    __shared__ int   sTok[TM];
    __shared__ float sWgt[TM];

    const int e   = blockIdx.z;
    const int cnt = counts[slot * NEXP + e];
    const int m0  = blockIdx.y * TM;
    if (m0 >= cnt) return;                      // block-uniform exit
    const int n0  = blockIdx.x * TN;

    const int tid  = threadIdx.x;
    const int lane = tid & 31;
    const int wv   = tid >> 5;                  // 0..7
    const int wm   = wv & 1;                    // row half (64 rows)
    const int wn   = wv >> 1;                   // col quarter (32 cols)

    const int*   lst = lists + (size_t)(slot * NEXP + e) * NTOK;
    const float* wls = wl    + (size_t)(slot * NEXP + e) * NTOK;

    if (tid < TM) {
        const int m  = m0 + tid;
        const int ok = (m < cnt);
        sTok[tid] = ok ? lst[m] : 0;
        sWgt[tid] = ok ? wls[m] : 0.0f;
    }
    __syncthreads();

    const float* We = W + (size_t)e * HDIM * HDIM;

    // ---- per-thread async-copy descriptors (4 x b128 for A, 4 x b128 for B)
    unsigned aLds[2][4], bLds[2][4];            // LDS byte offsets per buffer
    unsigned aVoff[4], bVoff[4];                // global byte offsets (k0 = 0)
    #pragma unroll
    for (int it = 0; it < 4; ++it) {
        const int q    = tid + it * 256;        // 0..1023
        const int arow = q >> 3;
        const int ac4  = (q & 7) << 2;
        const int brow = q >> 5;
        const int bc4  = (q & 31) << 2;
        aVoff[it] = ((unsigned)sTok[arow] * HDIM + (unsigned)ac4) * 4u;
        bVoff[it] = ((unsigned)brow * HDIM + (unsigned)(n0 + bc4)) * 4u;
        #pragma unroll
        for (int bu = 0; bu < 2; ++bu) {
            aLds[bu][it] = (unsigned)(size_t)&sA[bu][arow * LDA + ac4];
            bLds[bu][it] = (unsigned)(size_t)&sB[bu][brow * TN + bc4];
        }
    }

    v8f acc[4][2];
    #pragma unroll
    for (int i = 0; i < 4; ++i)
        #pragma unroll
        for (int j = 0; j < 2; ++j) acc[i][j] = (v8f){};

    const int lm = lane & 15;
    const int kh = (lane >> 4) << 1;            // 0 or 2 (A/B K-pair select)

    // prologue: start chunk 0 into buffer 0
    #pragma unroll
    for (int it = 0; it < 4; ++it) ASYNC_B128(aLds[0][it], aVoff[it], x);
    #pragma unroll
    for (int it = 0; it < 4; ++it) ASYNC_B128(bLds[0][it], bVoff[it], We);

    int buf = 0;
    for (int k0 = 0; k0 < HDIM; k0 += TK) {
        if (k0 + TK < HDIM) {
            // start next chunk into the other buffer (overlaps with compute)
            const unsigned aoff = (unsigned)((k0 + TK) * 4);
            const unsigned boff = (unsigned)((k0 + TK) * HDIM * 4);
            const int nb = buf ^ 1;
            #pragma unroll
            for (int it = 0; it < 4; ++it) ASYNC_B128(aLds[nb][it], aVoff[it] + aoff, x);
            #pragma unroll
            for (int it = 0; it < 4; ++it) ASYNC_B128(bLds[nb][it], bVoff[it] + boff, We);
            // in-order retirement: <=8 outstanding -> current chunk resident
            asm volatile("s_wait_asynccnt 8" ::: "memory");
        } else {
            asm volatile("s_wait_asynccnt 0" ::: "memory");
        }
        __syncthreads();                         // chunk visible to all waves

        const float* cA = sA[buf];
        const float* cB = sB[buf];

        // ---- 8 K-steps of V_WMMA_F32_16X16X4_F32
        #pragma unroll
        for (int kk = 0; kk < TK; kk += 4) {
            v2f afr[4], bfr[2];
            #pragma unroll
            for (int i = 0; i < 4; ++i) {
                const float* p = &cA[(wm * 64 + i * 16 + lm) * LDA + kk + kh];
                afr[i][0] = p[0]; afr[i][1] = p[1];
            }
            #pragma unroll
            for (int j = 0; j < 2; ++j) {
                const int col = wn * 32 + j * 16 + lm;
                bfr[j][0] = cB[(kk + kh)     * TN + col];
                bfr[j][1] = cB[(kk + kh + 1) * TN + col];
            }
            #pragma unroll
            for (int i = 0; i < 4; ++i)
                #pragma unroll
                for (int j = 0; j < 2; ++j)
                    acc[i][j] = __builtin_amdgcn_wmma_f32_16x16x4_f32(
                        false, afr[i], false, bfr[j],
                        (short)0, acc[i][j], false, false);
        }
        __syncthreads();                         // all waves done reading buf
        buf ^= 1;
    }

    // ---- epilogue: weighted scatter (unique writer per element in this pass)
    const float* be = b + (size_t)e * HDIM;
    const int rh = (lane >> 4) << 3;            // row offset 0 or 8 within 16-tile
    #pragma unroll
    for (int i = 0; i < 4; ++i) {
        #pragma unroll
        for (int j = 0; j < 2; ++j) {
            const int gn = n0 + wn * 32 + j * 16 + lm;
            const float bv = be[gn];
            #pragma unroll
            for (int r = 0; r < 8; ++r) {
                const int m = wm * 64 + i * 16 + rh + r;
                if (m0 + m < cnt) {
                    const float val = sWgt[m] * (acc[i][j][r] + bv);
                    float* po = &out[(size_t)sTok[m] * HDIM + gn];
                    if (slot == 0) *po = val;
                    else           *po += val;
                }
            }
        }
    }
}

// ---------------------------------------------------------------- launch
extern "C" void kernel_launch(void* const* d_in, const int* in_sizes, int n_in,
                              void* d_out, int out_size, void* d_ws, size_t ws_size,
                              hipStream_t stream) {
    const float* x  = (const float*)d_in[0];
    const float* Wg = (const float*)d_in[1];
    const float* bg = (const float*)d_in[2];
    const float* W  = (const float*)d_in[3];
    const float* b  = (const float*)d_in[4];
    (void)in_sizes; (void)n_in; (void)out_size; (void)ws_size;

    float* out        = (float*)d_out;                           // [T, H]
    float* logits_out = (float*)d_out + (size_t)NTOK * HDIM;     // [T, E]

    char* ws = (char*)d_ws;
    int*   counts = (int*)ws;                                    // [2*E]
    int*   lists  = (int*)(ws + 256);                            // [2*E][T]
    float* wl     = (float*)(ws + 256 + (size_t)2 * NEXP * NTOK * 4);

    zero_counts_kernel<<<1, 32, 0, stream>>>(counts);

    moe_router_kernel<<<NTOK / 32, 256, 0, stream>>>(
        x, Wg, bg, logits_out, counts, lists, wl);

    dim3 grid(HDIM / TN, NTOK / TM, NEXP);                       // (16, 64, 8)
    moe_expert_gemm_kernel<<<grid, 256, 0, stream>>>(
        x, W, b, counts, lists, wl, out, 0);
    moe_expert_gemm_kernel<<<grid, 256, 0, stream>>>(
        x, W, b, counts, lists, wl, out, 1);
}